// EdgePredictor_84121229459823
// MI455X (gfx1250) — compile-verified
//
#include <hip/hip_runtime.h>
#include <hip/hip_bf16.h>
#include <math.h>

typedef __attribute__((ext_vector_type(2))) float v2f;
typedef __attribute__((ext_vector_type(8))) float v8f;
typedef __attribute__((ext_vector_type(4))) int   v4i_t;

#define NN   4096
#define HH   64
#define CC   3
#define KNN  15
#define NEG_INF (-__builtin_huge_valf())

#define LDA 36   // LDS row stride (floats): 16B-aligned chunks + bank skew
#define LDB 36

// ---- gfx1250 async global->LDS path (guarded; fallback = sync staging) ----
#if defined(__HIP_DEVICE_COMPILE__) && __has_builtin(__builtin_amdgcn_global_load_async_to_lds_b128)
#define ASYNC_LDS 1
#else
#define ASYNC_LDS 0
#endif

#if ASYNC_LDS
typedef __attribute__((address_space(1))) v4i_t* gptr128;
typedef __attribute__((address_space(3))) v4i_t* lptr128;
__device__ __forceinline__ void gld_async_b128(const float* g, float* l) {
  __builtin_amdgcn_global_load_async_to_lds_b128((gptr128)(g), (lptr128)(l), 0, 0);
}
__device__ __forceinline__ void wait_async0() {
#if __has_builtin(__builtin_amdgcn_s_wait_asynccnt)
  __builtin_amdgcn_s_wait_asynccnt(0);
#else
  asm volatile("s_wait_asynccnt 0x0" ::: "memory");
#endif
}
#endif

// ---------------- atomic helpers ----------------
__device__ __forceinline__ void atomic_max_f32(float* addr, float val) {
  int* ia = (int*)addr;
  int old = *ia;
  while (__int_as_float(old) < val) {
    int assumed = old;
    old = atomicCAS(ia, assumed, __float_as_int(val));
    if (old == assumed) break;
  }
}

// ---------------- init ----------------
__global__ void init_f32(float* __restrict__ p, int n, float v) {
  int i = blockIdx.x * blockDim.x + threadIdx.x;
  if (i < n) p[i] = v;
}

// ---------------- brute-force kNN (one wave per node) ----------------
__global__ __launch_bounds__(32) void knn_kernel(const float* __restrict__ x,
                                                 int* __restrict__ knn_idx) {
  __shared__ float cd[32 * KNN];
  __shared__ int   ci[32 * KNN];
  const int i = blockIdx.x;
  const int lane = threadIdx.x;
  const float xi0 = x[i * 3 + 0], xi1 = x[i * 3 + 1], xi2 = x[i * 3 + 2];

  float bd[KNN];
  int   bi[KNN];
#pragma unroll
  for (int s = 0; s < KNN; ++s) { bd[s] = 3.4e38f; bi[s] = 0; }

  for (int j = lane; j < NN; j += 32) {
    if (j == i) continue;
    float d0 = xi0 - x[j * 3 + 0];
    float d1 = xi1 - x[j * 3 + 1];
    float d2 = xi2 - x[j * 3 + 2];
    float d = d0 * d0 + d1 * d1 + d2 * d2;
    if (d < bd[KNN - 1]) {
      int s = KNN - 1;
      while (s > 0 && bd[s - 1] > d) { bd[s] = bd[s - 1]; bi[s] = bi[s - 1]; --s; }
      bd[s] = d; bi[s] = j;
    }
  }
#pragma unroll
  for (int s = 0; s < KNN; ++s) { cd[lane * KNN + s] = bd[s]; ci[lane * KNN + s] = bi[s]; }
  __syncthreads();
  if (lane == 0) {
    for (int s = 0; s < KNN; ++s) {
      float m = 3.4e38f; int mi = 0, mp = -1;
      for (int t = 0; t < 32 * KNN; ++t)
        if (cd[t] < m) { m = cd[t]; mi = ci[t]; mp = t; }
      if (mp >= 0) cd[mp] = 3.4e38f;
      knn_idx[i * KNN + s] = mi;
    }
  }
}

// ---------------- DevConv edge pass: segment-max of W_theta(x_r - x_c)+b ----------------
__global__ void devconv_kernel(const float* __restrict__ x, const int* __restrict__ ei, int E,
                               const int* __restrict__ knn_idx,
                               const float* __restrict__ Wt, const float* __restrict__ bt,
                               float* __restrict__ agg, int Eext) {
  int tid = blockIdx.x * blockDim.x + threadIdx.x;
  int e = tid >> 6, j = tid & 63;
  if (e >= Eext) return;
  int r, c;
  if (e < E) { r = ei[e]; c = ei[E + e]; }
  else { int t = e - E; r = t / KNN; c = knn_idx[t]; }
  float d0 = x[r * 3 + 0] - x[c * 3 + 0];
  float d1 = x[r * 3 + 1] - x[c * 3 + 1];
  float d2 = x[r * 3 + 2] - x[c * 3 + 2];
  float h = Wt[j * 3 + 0] * d0 + Wt[j * 3 + 1] * d1 + Wt[j * 3 + 2] * d2 + bt[j];
  atomic_max_f32(&agg[r * HH + j], h);
}

__global__ void clamp_kernel(float* __restrict__ p, int n) {
  int i = blockIdx.x * blockDim.x + threadIdx.x;
  if (i < n) { float v = p[i]; p[i] = (v < -1.0e37f || !(v == v)) ? 0.0f : v; }
}

// ---------------- small GEMM: Out[4096x64] = In[4096x64] * W^T + bias, fp32 WMMA ----------------
__global__ __launch_bounds__(128) void gemm_h64(const float* __restrict__ In,
                                                const float* __restrict__ W,
                                                const float* __restrict__ bias,
                                                float* __restrict__ Out) {
  int wid  = (blockIdx.x * blockDim.x + threadIdx.x) >> 5;  // global wave id, 0..1023
  int lane = threadIdx.x & 31;
  int mt = wid >> 2;          // 0..255  (M tile)
  int nt = wid & 3;           // 0..3    (N tile)
  int h = lane >> 4, t = lane & 15;
  int row  = mt * 16 + t;
  int coln = nt * 16 + t;

  v8f acc = {0.f, 0.f, 0.f, 0.f, 0.f, 0.f, 0.f, 0.f};
#pragma unroll
  for (int k = 0; k < HH; k += 4) {
    v2f a, b;
    a.x = In[row * HH + k + 2 * h];
    a.y = In[row * HH + k + 2 * h + 1];
    b.x = W[coln * HH + k + 2 * h];        // B[k][n] = W[n][k]  (W^T)
    b.y = W[coln * HH + k + 2 * h + 1];
    acc = __builtin_amdgcn_wmma_f32_16x16x4_f32(false, a, false, b, (short)0, acc, false, false);
  }
  float bb = bias ? bias[coln] : 0.0f;
#pragma unroll
  for (int v = 0; v < 8; ++v) {
    int orow = mt * 16 + v + 8 * h;
    Out[orow * HH + coln] = acc[v] + bb;
  }
}

// ---------------- attention logits per edge ----------------
__global__ void attn_kernel(const float* __restrict__ q, const float* __restrict__ kf,
                            const int* __restrict__ ei, int E, float* __restrict__ attn) {
  int e = blockIdx.x * blockDim.x + threadIdx.x;
  if (e >= E) return;
  int r = ei[e], c = ei[E + e];
  const float4* qr = (const float4*)(q + (size_t)r * HH);
  const float4* kc = (const float4*)(kf + (size_t)c * HH);
  float s = 0.f;
#pragma unroll
  for (int t = 0; t < 16; ++t) {
    float4 a = qr[t], b = kc[t];
    s += a.x * b.x + a.y * b.y + a.z * b.z + a.w * b.w;
  }
  attn[e] = s;
}

__global__ void rowmax_kernel(const float* __restrict__ attn, const int* __restrict__ ei, int E,
                              float* __restrict__ rowmax) {
  int e = blockIdx.x * blockDim.x + threadIdx.x;
  if (e >= E) return;
  atomic_max_f32(&rowmax[ei[e]], attn[e]);
}

__global__ void expsum_kernel(const float* __restrict__ attn, const int* __restrict__ ei, int E,
                              const float* __restrict__ rowmax, float* __restrict__ rowsum,
                              float* __restrict__ scores) {
  int e = blockIdx.x * blockDim.x + threadIdx.x;
  if (e >= E) return;
  int r = ei[e];
  float m = rowmax[r];
  if (m < -1.0e37f || !(m == m)) m = 0.0f;
  float ev = __expf(attn[e] - m);
  scores[e] = ev;
  atomicAdd(&rowsum[r], ev);
}

__global__ void scatter_kernel(const float* __restrict__ scores, const float* __restrict__ rowsum,
                               const int* __restrict__ ei, int E,
                               float* __restrict__ S, float* __restrict__ A) {
  int e = blockIdx.x * blockDim.x + threadIdx.x;
  if (e >= E) return;
  int r = ei[e], c = ei[E + e];
  float sc = scores[e] / rowsum[r];
  atomicAdd(&S[(size_t)r * NN + c], sc);
  atomicAdd(&A[(size_t)r * NN + c], 1.0f);
}

// ---------------- big GEMM: C[4096x4096] = Ag * Bg (BT=0) or Ag * Bg^T (BT=1) ----------------
// Block = 256 threads = 8 waves (4 M x 2 N), block tile 64x32, K-stage 32, fp32 WMMA.
// LDS layouts: sA[m][k] (stride LDA). sB: BT=0 -> [k][n], BT=1 -> [n][k] (stride LDB):
// global reads and 16B LDS chunks are contiguous in both cases.
template <bool BT>
__global__ __launch_bounds__(256) void gemm4k(const float* __restrict__ Ag,
                                              const float* __restrict__ Bg,
                                              float* __restrict__ Cg) {
  const int tid  = threadIdx.x;
  const int wid  = tid >> 5, lane = tid & 31;
  const int wm = wid & 3, wn = wid >> 2;
  const int h = lane >> 4, t = lane & 15;
  const int rowA0 = blockIdx.x * 64;
  const int colB0 = blockIdx.y * 32;

  v8f acc = {0.f, 0.f, 0.f, 0.f, 0.f, 0.f, 0.f, 0.f};

#if ASYNC_LDS
  // -------- double-buffered async global->LDS pipeline --------
  __shared__ float sA[2][64 * LDA];
  __shared__ float sB[2][32 * LDB];

  // A tile: 512 float4 chunks (2/thread); B tile: 256 float4 chunks (1/thread)
  const int ca0_m = tid >> 3,          ca0_k = (tid & 7) * 4;          // chunk tid
  const int ca1_m = (tid + 256) >> 3,  ca1_k = (tid & 7) * 4;          // chunk tid+256
  const int cb_r  = tid >> 3,          cb_k  = (tid & 7) * 4;          // B chunk (r = k-row or n-row)

  auto issue = [&](int ks, int buf) {
    gld_async_b128(&Ag[(size_t)(rowA0 + ca0_m) * NN + ks + ca0_k], &sA[buf][ca0_m * LDA + ca0_k]);
    gld_async_b128(&Ag[(size_t)(rowA0 + ca1_m) * NN + ks + ca1_k], &sA[buf][ca1_m * LDA + ca1_k]);
    if (BT) {   // sB[n][k] = Bg[colB0+n][ks+k]
      gld_async_b128(&Bg[(size_t)(colB0 + cb_r) * NN + ks + cb_k], &sB[buf][cb_r * LDB + cb_k]);
    } else {    // sB[k][n] = Bg[ks+k][colB0+n]
      gld_async_b128(&Bg[(size_t)(ks + cb_r) * NN + colB0 + cb_k], &sB[buf][cb_r * LDB + cb_k]);
    }
  };

  int buf = 0;
  issue(0, buf);
  for (int ks = 0; ks < NN; ks += 32) {
    wait_async0();
    __syncthreads();                       // current buffer fully landed for all waves
    if (ks + 32 < NN) issue(ks + 32, buf ^ 1);   // overlap next-stage copy with compute
#pragma unroll
    for (int k0 = 0; k0 < 32; k0 += 4) {
      v2f a, b;
      a.x = sA[buf][(wm * 16 + t) * LDA + k0 + 2 * h];
      a.y = sA[buf][(wm * 16 + t) * LDA + k0 + 2 * h + 1];
      if (BT) {
        b.x = sB[buf][(wn * 16 + t) * LDB + k0 + 2 * h];
        b.y = sB[buf][(wn * 16 + t) * LDB + k0 + 2 * h + 1];
      } else {
        b.x = sB[buf][(k0 + 2 * h) * LDB + wn * 16 + t];
        b.y = sB[buf][(k0 + 2 * h + 1) * LDB + wn * 16 + t];
      }
      acc = __builtin_amdgcn_wmma_f32_16x16x4_f32(false, a, false, b, (short)0, acc, false, false);
    }
    __syncthreads();                       // all waves done reading buf before it is refilled
    buf ^= 1;
  }
#else
  // -------- synchronous staging fallback --------
  __shared__ float sA[64 * LDA];
  __shared__ float sB[32 * LDB];
  for (int ks = 0; ks < NN; ks += 32) {
    if (ks + 32 < NN) {
      __builtin_prefetch(&Ag[(size_t)(rowA0 + (tid >> 2)) * NN + ks + 32], 0, 1);
      if (BT)
        __builtin_prefetch(&Bg[(size_t)(colB0 + (tid & 31)) * NN + ks + 32], 0, 1);
      else
        __builtin_prefetch(&Bg[(size_t)(ks + 32 + (tid >> 3)) * NN + colB0], 0, 1);
    }
#pragma unroll
    for (int it = 0; it < 8; ++it) {
      int li = tid + it * 256;
      int m = li >> 5, k = li & 31;
      sA[m * LDA + k] = Ag[(size_t)(rowA0 + m) * NN + ks + k];
    }
#pragma unroll
    for (int it = 0; it < 4; ++it) {
      int li = tid + it * 256;
      int r = li >> 5, c = li & 31;
      if (BT) sB[r * LDB + c] = Bg[(size_t)(colB0 + r) * NN + ks + c];      // [n][k]
      else    sB[r * LDB + c] = Bg[(size_t)(ks + r) * NN + colB0 + c];      // [k][n]
    }
    __syncthreads();
#pragma unroll
    for (int k0 = 0; k0 < 32; k0 += 4) {
      v2f a, b;
      a.x = sA[(wm * 16 + t) * LDA + k0 + 2 * h];
      a.y = sA[(wm * 16 + t) * LDA + k0 + 2 * h + 1];
      if (BT) {
        b.x = sB[(wn * 16 + t) * LDB + k0 + 2 * h];
        b.y = sB[(wn * 16 + t) * LDB + k0 + 2 * h + 1];
      } else {
        b.x = sB[(k0 + 2 * h) * LDB + wn * 16 + t];
        b.y = sB[(k0 + 2 * h + 1) * LDB + wn * 16 + t];
      }
      acc = __builtin_amdgcn_wmma_f32_16x16x4_f32(false, a, false, b, (short)0, acc, false, false);
    }
    __syncthreads();
  }
#endif

#pragma unroll
  for (int v = 0; v < 8; ++v) {
    int orow = rowA0 + wm * 16 + v + 8 * h;
    int ocol = colB0 + wn * 16 + t;
    Cg[(size_t)orow * NN + ocol] = acc[v];
  }
}

// ---------------- launcher ----------------
extern "C" void kernel_launch(void* const* d_in, const int* in_sizes, int n_in,
                              void* d_out, int out_size, void* d_ws, size_t ws_size,
                              hipStream_t stream) {
  const float* x    = (const float*)d_in[0];
  const int*   ei   = (const int*)d_in[1];
  const float* Wt   = (const float*)d_in[2];
  const float* bt   = (const float*)d_in[3];
  const float* Wphi = (const float*)d_in[4];
  const float* bphi = (const float*)d_in[5];
  const float* Wq   = (const float*)d_in[6];
  const float* Wk   = (const float*)d_in[7];

  const int E    = in_sizes[1] / 2;
  const int Eext = E + NN * KNN;

  float* ws    = (float*)d_ws;
  float* S     = ws;                           // 16.7M
  float* A     = S + (size_t)NN * NN;          // 16.7M
  float* T     = A + (size_t)NN * NN;          // 16.7M
  float* agg   = T + (size_t)NN * NN;          // N*H
  float* feats = agg + NN * HH;
  float* q     = feats + NN * HH;
  float* kf    = q + NN * HH;
  float* attn  = kf + NN * HH;                 // E
  float* scor  = attn + E;                     // E
  float* rmax  = scor + E;                     // N
  float* rsum  = rmax + NN;                    // N
  int*   knni  = (int*)(rsum + NN);            // N*K ints

  // zero dense S and A (contiguous), init reductions
  (void)hipMemsetAsync(S, 0, (size_t)2 * NN * NN * sizeof(float), stream);
  init_f32<<<(NN * HH + 255) / 256, 256, 0, stream>>>(agg, NN * HH, NEG_INF);
  init_f32<<<(NN + 255) / 256, 256, 0, stream>>>(rmax, NN, NEG_INF);
  init_f32<<<(NN + 255) / 256, 256, 0, stream>>>(rsum, NN, 0.0f);

  // 1) kNN graph
  knn_kernel<<<NN, 32, 0, stream>>>(x, knni);

  // 2) DevConv over extended edges (segment-max), clamp, W_phi projection
  {
    long long tot = (long long)Eext * HH;
    devconv_kernel<<<(int)((tot + 255) / 256), 256, 0, stream>>>(x, ei, E, knni, Wt, bt, agg, Eext);
  }
  clamp_kernel<<<(NN * HH + 255) / 256, 256, 0, stream>>>(agg, NN * HH);
  gemm_h64<<<256, 128, 0, stream>>>(agg, Wphi, bphi, feats);

  // 3) q/k projections (WMMA)
  gemm_h64<<<256, 128, 0, stream>>>(feats, Wq, nullptr, q);
  gemm_h64<<<256, 128, 0, stream>>>(feats, Wk, nullptr, kf);

  // 4) edge attention + scatter softmax
  attn_kernel<<<(E + 255) / 256, 256, 0, stream>>>(q, kf, ei, E, attn);
  rowmax_kernel<<<(E + 255) / 256, 256, 0, stream>>>(attn, ei, E, rmax);
  expsum_kernel<<<(E + 255) / 256, 256, 0, stream>>>(attn, ei, E, rmax, rsum, scor);

  // 5) dense S, A scatter
  scatter_kernel<<<(E + 255) / 256, 256, 0, stream>>>(scor, rsum, ei, E, S, A);

  // 6) A_s = (S @ A) @ S^T  — L2-resident fp32 WMMA GEMMs
  dim3 g(NN / 64, NN / 32);
  gemm4k<false><<<g, 256, 0, stream>>>(S, A, T);
  gemm4k<true><<<g, 256, 0, stream>>>(T, S, (float*)d_out);
}